// MambaStackedTest_79182017069441
// MI455X (gfx1250) — compile-verified
//
#include <hip/hip_runtime.h>
#include <hip/hip_bf16.h>
#include <cstddef>

// ---- problem constants (from reference) ----
#define NBLK     3
#define DMODEL   512
#define DINNER   1024
#define DSTATE   8
#define KCONV    4
#define LSEQ     2048
#define NBATCH   4
#define NCLS_    10
#define DTRANK   32
#define DBL_W    (DTRANK + 2*DSTATE)   // 48

typedef __attribute__((ext_vector_type(2))) float v2f;
typedef __attribute__((ext_vector_type(4))) float v4f;
typedef __attribute__((ext_vector_type(8))) float v8f;

__device__ __forceinline__ float softplusf(float v) {
    return fmaxf(v, 0.f) + log1pf(__expf(-fabsf(v)));
}
__device__ __forceinline__ float siluf(float v) {
    return v / (1.f + __expf(-v));
}
__device__ __forceinline__ v8f wmma4(v2f a, v2f b, v8f c) {
    return __builtin_amdgcn_wmma_f32_16x16x4_f32(false, a, false, b, (short)0, c, false, false);
}

// ---------------------------------------------------------------------------
// LDS-tiled fp32 WMMA GEMM: C[M,N] = act(A@B + bias) (+ resid)
// Block: 256 thr (8 waves) -> 128x128 tile; waves 4(M: 32 rows) x 2(N: 64 cols).
// K chunked by 32, double-buffered in LDS; prefetch overlaps 64 WMMAs/chunk.
// B staged k-pair interleaved so B fragments are single aligned ds_load_b64.
// Requires: M%128==0, N%128==0, K%32==0, lda/ldb %4==0, 16B-aligned bases.
// ---------------------------------------------------------------------------
#define KC         32
#define BM         128
#define BN         128
#define AS_STRIDE  36    // floats per A row: 144B = 16B aligned, bank-spread
#define BS_PSTRIDE 288   // floats per B k-pair row: disjoint bank windows

template<int ACT, bool RESID>
__global__ __launch_bounds__(256)
void gemm_wmma_lds(const float* __restrict__ A, const float* __restrict__ B,
                   const float* __restrict__ bias, const float* __restrict__ resid,
                   float* __restrict__ C,
                   int K, int lda, int ldb, int ldc, int ldr) {
    __shared__ float As[2][BM * AS_STRIDE];          // 36,864 B
    __shared__ float Bs[2][(KC/2) * BS_PSTRIDE];     // 36,864 B

    const int tid  = threadIdx.x;
    const int lane = tid & 31;
    const int wave = tid >> 5;
    const int wm   = wave >> 1;      // 0..3 -> 32-row M sub-tile
    const int wn   = wave & 1;       // 0..1 -> 64-col N sub-tile
    const int m0   = blockIdx.y * BM;
    const int n0   = blockIdx.x * BN;

    // --- per-thread copy slots ---
    // A tile: 128x32 = 1024 v4f -> 4 per thread (row-major)
    // B tile: 32x128 -> 1024 groups of (k-pair p, col-pair cp); 4 per thread
    int aRow[4], aCol[4], bPair[4], bCp[4];
    #pragma unroll
    for (int i = 0; i < 4; ++i) {
        int e = tid + i*256;
        aRow[i] = e >> 3;  aCol[i] = (e & 7) * 4;
        bPair[i] = e >> 6; bCp[i]  = e & 63;
    }

    const float* aP[4]; const float* bP[4];
    #pragma unroll
    for (int i = 0; i < 4; ++i) aP[i] = A + (size_t)(m0 + aRow[i]) * lda + aCol[i];
    #pragma unroll
    for (int i = 0; i < 4; ++i) bP[i] = B + (size_t)(2*bPair[i]) * ldb + n0 + 2*bCp[i];
    const size_t bStep = (size_t)KC * ldb;

    // --- prologue: stage chunk 0 into buffer 0 ---
    {
        v4f ta[4]; v2f tb0[4], tb1[4];
        #pragma unroll
        for (int i = 0; i < 4; ++i) ta[i]  = *(const v4f*)aP[i];
        #pragma unroll
        for (int i = 0; i < 4; ++i) { tb0[i] = *(const v2f*)bP[i];
                                      tb1[i] = *(const v2f*)(bP[i] + ldb); }
        #pragma unroll
        for (int i = 0; i < 4; ++i) *(v4f*)&As[0][aRow[i]*AS_STRIDE + aCol[i]] = ta[i];
        #pragma unroll
        for (int i = 0; i < 4; ++i) {
            v4f w; w.x = tb0[i].x; w.y = tb1[i].x; w.z = tb0[i].y; w.w = tb1[i].y;
            *(v4f*)&Bs[0][bPair[i]*BS_PSTRIDE + bCp[i]*4] = w;
        }
    }
    __syncthreads();

    v8f acc[2][4];
    #pragma unroll
    for (int s = 0; s < 2; ++s)
        #pragma unroll
        for (int i = 0; i < 4; ++i)
            #pragma unroll
            for (int r = 0; r < 8; ++r) acc[s][i][r] = 0.f;

    const int rowA  = lane & 15;
    const int khalf = (lane >> 4) << 1;          // 0 or 2
    const int pHalf = khalf >> 1;                // 0 or 1
    const int nChunks = K / KC;

    for (int c = 0; c < nChunks; ++c) {
        const int cur  = c & 1;
        const bool more = (c + 1 < nChunks);

        // prefetch next chunk into VGPRs (overlapped with the WMMAs below)
        v4f ta[4]; v2f tb0[4], tb1[4];
        if (more) {
            #pragma unroll
            for (int i = 0; i < 4; ++i) ta[i]  = *(const v4f*)(aP[i] + KC);
            #pragma unroll
            for (int i = 0; i < 4; ++i) { tb0[i] = *(const v2f*)(bP[i] + bStep);
                                          tb1[i] = *(const v2f*)(bP[i] + bStep + ldb); }
        }

        const float* __restrict__ as0 = &As[cur][(wm*32 + rowA) * AS_STRIDE + khalf];
        const float* __restrict__ as1 = as0 + 16 * AS_STRIDE;
        const float* __restrict__ bsb = &Bs[cur][pHalf * BS_PSTRIDE + (wn*64 + rowA) * 2];
        #pragma unroll
        for (int k0 = 0; k0 < KC; k0 += 4) {
            v2f a0 = *(const v2f*)(as0 + k0);
            v2f a1 = *(const v2f*)(as1 + k0);
            const float* __restrict__ bk = bsb + (k0 >> 1) * BS_PSTRIDE;
            #pragma unroll
            for (int i = 0; i < 4; ++i) {
                v2f b = *(const v2f*)(bk + i*32);   // single ds_load_b64
                acc[0][i] = wmma4(a0, b, acc[0][i]);
                acc[1][i] = wmma4(a1, b, acc[1][i]);
            }
        }

        if (more) {
            const int nxt = cur ^ 1;
            #pragma unroll
            for (int i = 0; i < 4; ++i) *(v4f*)&As[nxt][aRow[i]*AS_STRIDE + aCol[i]] = ta[i];
            #pragma unroll
            for (int i = 0; i < 4; ++i) {
                v4f w; w.x = tb0[i].x; w.y = tb1[i].x; w.z = tb0[i].y; w.w = tb1[i].y;
                *(v4f*)&Bs[nxt][bPair[i]*BS_PSTRIDE + bCp[i]*4] = w;
            }
            #pragma unroll
            for (int i = 0; i < 4; ++i) aP[i] += KC;
            #pragma unroll
            for (int i = 0; i < 4; ++i) bP[i] += bStep;
            __syncthreads();
        }
    }

    // --- epilogue: C/D layout -> row = r + 8*(lane>>4), col = lane&15 ---
    #pragma unroll
    for (int s = 0; s < 2; ++s) {
        const int crow0 = m0 + wm*32 + s*16 + ((lane >> 4) << 3);
        #pragma unroll
        for (int i = 0; i < 4; ++i) {
            const int ccol = n0 + wn*64 + i*16 + (lane & 15);
            const float bv = bias ? bias[ccol] : 0.f;
            #pragma unroll
            for (int r = 0; r < 8; ++r) {
                float v = acc[s][i][r] + bv;
                if (ACT == 1) v = softplusf(v);
                const size_t row = (size_t)(crow0 + r);
                if (RESID) v += resid[row * ldr + ccol];
                C[row * ldc + ccol] = v;
            }
        }
    }
}

// ---------------------------------------------------------------------------
// Simple per-wave WMMA GEMM for the small N=48 x_proj (one 16x48 tile/wave).
// ---------------------------------------------------------------------------
template<int NT>
__global__ __launch_bounds__(256)
void gemm_wmma_nt(const float* __restrict__ A, const float* __restrict__ B,
                  float* __restrict__ C, int M, int N, int K,
                  int lda, int ldb, int ldc) {
    const int lane = threadIdx.x & 31;
    const int wave = threadIdx.x >> 5;
    const int tilesN = N / (16 * NT);
    const int tilesM = M / 16;
    const int wid = blockIdx.x * (blockDim.x >> 5) + wave;
    if (wid >= tilesM * tilesN) return;
    const int tm = wid / tilesN;
    const int tn = wid % tilesN;

    const int rowIn = lane & 15;
    const int khalf = (lane >> 4) << 1;
    const float* __restrict__ Arow = A + (size_t)(tm*16 + rowIn) * lda + khalf;
    const float* __restrict__ Bcol = B + (size_t)khalf * ldb + tn*(16*NT) + rowIn;

    v8f acc[NT];
    #pragma unroll
    for (int i = 0; i < NT; ++i)
        #pragma unroll
        for (int r = 0; r < 8; ++r) acc[i][r] = 0.f;

    for (int k0 = 0; k0 < K; k0 += 4) {
        v2f a = *(const v2f*)(Arow);
        Arow += 4;
        #pragma unroll
        for (int i = 0; i < NT; ++i) {
            v2f b;
            b.x = Bcol[i*16];
            b.y = Bcol[i*16 + ldb];
            acc[i] = wmma4(a, b, acc[i]);
        }
        Bcol += (size_t)4 * ldb;
    }

    const int crow0 = tm*16 + ((lane >> 4) << 3);
    #pragma unroll
    for (int i = 0; i < NT; ++i) {
        const int ccol = tn*(16*NT) + i*16 + (lane & 15);
        #pragma unroll
        for (int r = 0; r < 8; ++r)
            C[(size_t)(crow0 + r) * ldc + ccol] = acc[i][r];
    }
}

// ---------------------------------------------------------------------------
// Depthwise causal conv (K=4) over L on u = xz[..., :DINNER], + bias, SiLU.
// ---------------------------------------------------------------------------
__global__ __launch_bounds__(256)
void conv_silu_kernel(const float* __restrict__ xz, const float* __restrict__ cw,
                      const float* __restrict__ cb, float* __restrict__ u) {
    const int g = blockIdx.x * blockDim.x + threadIdx.x;
    if (g >= NBATCH * LSEQ * DINNER) return;
    const int d = g % DINNER;
    const int t = (g / DINNER) % LSEQ;
    const int b = g / (DINNER * LSEQ);

    float acc = cb[d];
    #pragma unroll
    for (int j = 0; j < KCONV; ++j) {
        const int tl = t - (KCONV - 1) + j;
        if (tl >= 0)
            acc += xz[((size_t)b * LSEQ + tl) * (2 * DINNER) + d] * cw[d * KCONV + j];
    }
    u[((size_t)b * LSEQ + t) * DINNER + d] = siluf(acc);
}

// ---------------------------------------------------------------------------
// Selective scan: one thread per (b, d). delta is read and overwritten with
// the fused output y = (scan + u*D) * silu(z).  B_t/C_t loaded as float4.
// ---------------------------------------------------------------------------
__global__ __launch_bounds__(64)
void scan_kernel(float* __restrict__ dy, const float* __restrict__ u,
                 const float* __restrict__ xz, const float* __restrict__ dbl,
                 const float* __restrict__ A_log, const float* __restrict__ Dp) {
    const int g = blockIdx.x * blockDim.x + threadIdx.x;
    if (g >= NBATCH * DINNER) return;
    const int d = g % DINNER;
    const int b = g / DINNER;

    float An[DSTATE], h[DSTATE];
    #pragma unroll
    for (int n = 0; n < DSTATE; ++n) {
        An[n] = -__expf(A_log[d * DSTATE + n]);
        h[n] = 0.f;
    }
    const float Dd = Dp[d];

    for (int t = 0; t < LSEQ; ++t) {
        const size_t bt = (size_t)b * LSEQ + t;
        const float dlt = dy[bt * DINNER + d];
        const float ut  = u[bt * DINNER + d];
        const float zt  = xz[bt * (2 * DINNER) + DINNER + d];
        const float* __restrict__ bl = dbl + bt * DBL_W;
        v4f Bt0 = *(const v4f*)(bl + DTRANK);
        v4f Bt1 = *(const v4f*)(bl + DTRANK + 4);
        v4f Ct0 = *(const v4f*)(bl + DTRANK + DSTATE);
        v4f Ct1 = *(const v4f*)(bl + DTRANK + DSTATE + 4);
        const float du = dlt * ut;
        float y = 0.f;
        #pragma unroll
        for (int n = 0; n < 4; ++n) {
            h[n] = __expf(dlt * An[n]) * h[n] + du * Bt0[n];
            y += h[n] * Ct0[n];
        }
        #pragma unroll
        for (int n = 0; n < 4; ++n) {
            h[4+n] = __expf(dlt * An[4+n]) * h[4+n] + du * Bt1[n];
            y += h[4+n] * Ct1[n];
        }
        dy[bt * DINNER + d] = (y + ut * Dd) * siluf(zt);
    }
}

// ---------------------------------------------------------------------------
// Mean pool over L, then tiny classifier.
// ---------------------------------------------------------------------------
__global__ __launch_bounds__(256)
void pool_kernel(const float* __restrict__ x, float* __restrict__ pooled) {
    const int g = blockIdx.x * blockDim.x + threadIdx.x;
    if (g >= NBATCH * DMODEL) return;
    const int dm = g % DMODEL;
    const int b  = g / DMODEL;
    float s = 0.f;
    for (int t = 0; t < LSEQ; ++t)
        s += x[((size_t)b * LSEQ + t) * DMODEL + dm];
    pooled[b * DMODEL + dm] = s * (1.f / LSEQ);
}

__global__ __launch_bounds__(64)
void cls_kernel(const float* __restrict__ pooled, const float* __restrict__ w,
                const float* __restrict__ bias, float* __restrict__ out) {
    const int g = threadIdx.x;
    if (g >= NBATCH * NCLS_) return;
    const int c = g % NCLS_;
    const int b = g / NCLS_;
    float s = bias[c];
    for (int k = 0; k < DMODEL; ++k)
        s += pooled[b * DMODEL + k] * w[k * NCLS_ + c];
    out[b * NCLS_ + c] = s;
}

// ---------------------------------------------------------------------------
extern "C" void kernel_launch(void* const* d_in, const int* in_sizes, int n_in,
                              void* d_out, int out_size, void* d_ws, size_t ws_size,
                              hipStream_t stream) {
    const float* x_in    = (const float*)d_in[0];
    const float* Wi_all  = (const float*)d_in[1];
    const float* cw_all  = (const float*)d_in[2];
    const float* cb_all  = (const float*)d_in[3];
    const float* Wx_all  = (const float*)d_in[4];
    const float* Wdt_all = (const float*)d_in[5];
    const float* bdt_all = (const float*)d_in[6];
    const float* Alog_all= (const float*)d_in[7];
    const float* D_all   = (const float*)d_in[8];
    const float* Wo_all  = (const float*)d_in[9];
    const float* cls_w   = (const float*)d_in[10];
    const float* cls_b   = (const float*)d_in[11];
    float* out = (float*)d_out;

    const int Mrows = NBATCH * LSEQ;            // 8192

    float* ws     = (float*)d_ws;
    float* xz     = ws;                                          // B*L*2048
    float* u      = xz    + (size_t)Mrows * 2 * DINNER;          // B*L*1024
    float* dbl    = u     + (size_t)Mrows * DINNER;              // B*L*48
    float* delta  = dbl   + (size_t)Mrows * DBL_W;               // B*L*1024 (becomes y)
    float* xb0    = delta + (size_t)Mrows * DINNER;              // B*L*512
    float* xb1    = xb0   + (size_t)Mrows * DMODEL;              // B*L*512
    float* pooled = xb1   + (size_t)Mrows * DMODEL;              // B*512

    const float* xcur = x_in;
    for (int i = 0; i < NBLK; ++i) {
        const float* Wi  = Wi_all  + (size_t)i * DMODEL * 2 * DINNER;
        const float* cw  = cw_all  + (size_t)i * DINNER * KCONV;
        const float* cb  = cb_all  + (size_t)i * DINNER;
        const float* Wx  = Wx_all  + (size_t)i * DINNER * DBL_W;
        const float* Wdt = Wdt_all + (size_t)i * DTRANK * DINNER;
        const float* bdt = bdt_all + (size_t)i * DINNER;
        const float* Alg = Alog_all+ (size_t)i * DINNER * DSTATE;
        const float* Dp  = D_all   + (size_t)i * DINNER;
        const float* Wo  = Wo_all  + (size_t)i * DINNER * DMODEL;
        float* xnext = (i & 1) ? xb1 : xb0;

        // 1) xz = x @ Wi : M=8192 N=2048 K=512
        {
            dim3 grid((2 * DINNER) / BN, Mrows / BM);
            gemm_wmma_lds<0, false><<<grid, 256, 0, stream>>>(
                xcur, Wi, nullptr, nullptr, xz,
                DMODEL, DMODEL, 2 * DINNER, 2 * DINNER, 0);
        }
        // 2) depthwise causal conv + bias + silu -> u
        {
            int total = NBATCH * LSEQ * DINNER;
            conv_silu_kernel<<<(total + 255) / 256, 256, 0, stream>>>(xz, cw, cb, u);
        }
        // 3) dbl = u @ Wx : M=8192 N=48 K=1024 (small N -> per-wave kernel)
        {
            int waves = (Mrows / 16) * (DBL_W / 48);
            gemm_wmma_nt<3><<<(waves + 7) / 8, 256, 0, stream>>>(
                u, Wx, dbl, Mrows, DBL_W, DINNER, DINNER, DBL_W, DBL_W);
        }
        // 4) delta = softplus(dt @ Wdt + bdt) : M=8192 N=1024 K=32
        {
            dim3 grid(DINNER / BN, Mrows / BM);
            gemm_wmma_lds<1, false><<<grid, 256, 0, stream>>>(
                dbl, Wdt, bdt, nullptr, delta,
                DTRANK, DBL_W, DINNER, DINNER, 0);
        }
        // 5) selective scan (delta overwritten with fused y)
        {
            int total = NBATCH * DINNER;
            scan_kernel<<<(total + 63) / 64, 64, 0, stream>>>(delta, u, xz, dbl, Alg, Dp);
        }
        // 6) x = x + y @ Wo : M=8192 N=512 K=1024
        {
            dim3 grid(DMODEL / BN, Mrows / BM);
            gemm_wmma_lds<0, true><<<grid, 256, 0, stream>>>(
                delta, Wo, nullptr, xcur, xnext,
                DINNER, DINNER, DMODEL, DMODEL, DMODEL);
        }
        xcur = xnext;
    }

    // 7) mean pool + classifier
    {
        int total = NBATCH * DMODEL;
        pool_kernel<<<(total + 255) / 256, 256, 0, stream>>>(xcur, pooled);
        cls_kernel<<<1, 64, 0, stream>>>(pooled, cls_w, cls_b, out);
    }
}